// TextBiLSTM_68719477196
// MI455X (gfx1250) — compile-verified
//
#include <hip/hip_runtime.h>
#include <hip/hip_bf16.h>

typedef __attribute__((ext_vector_type(2))) float v2f;
typedef __attribute__((ext_vector_type(8))) float v8f;

#define VOCAB 50257
#define EMB   128
#define HID   256
#define NOUT  32
#define BATCH 128
#define SEQ   2048
#define XPAD  132   // padded LDS row stride: 132 % 64 = 4 -> conflict-free column reads

// CDNA5 has a hardware transcendental V_TANH_F32 (gfx1250). Use it if the
// toolchain declares the builtin; fall back to libm otherwise.
__device__ __forceinline__ float fast_tanh(float x) {
#if __has_builtin(__builtin_amdgcn_tanhf)
    return __builtin_amdgcn_tanhf(x);
#else
    return tanhf(x);
#endif
}

// sigmoid(x) = 0.5*tanh(x/2) + 0.5  -> single TRANS op + 2 VALU, no rcp/branches
__device__ __forceinline__ float sigmoidf_(float x) {
    return __builtin_fmaf(0.5f, fast_tanh(0.5f * x), 0.5f);
}

// ---------------------------------------------------------------------------
// Kernel 1: embedding gather (only t=0 and t=S-1 are live) + i/g/o gate GEMMs
// via V_WMMA_F32_16X16X4_F32 + LSTM nonlinearity -> hy[128][256] in workspace.
// Grid: 8 blocks x 256 threads (8 waves). Block b owns rows [16b, 16b+16).
// Wave w owns hy column tiles n0 = 32w and 32w+16.
// ---------------------------------------------------------------------------
__global__ __launch_bounds__(256) void bilstm_gates_kernel(
    const int*   __restrict__ inputs,
    const float* __restrict__ weight,
    const float* __restrict__ Wxf, const float* __restrict__ bxf, const float* __restrict__ bhf,
    const float* __restrict__ Wxb, const float* __restrict__ bxb, const float* __restrict__ bhb,
    float* __restrict__ hy)
{
    __shared__ __align__(16) float Xs[2][16][XPAD];  // [dir][row][k]
    __shared__ int toks[2][16];

    const int tid    = threadIdx.x;
    const int m_base = blockIdx.x * 16;

    // token ids: dir 0 = forward (last timestep), dir 1 = backward (first)
    if (tid < 32) {
        const int mat = tid >> 4;
        const int row = tid & 15;
        const int t   = mat ? 0 : (SEQ - 1);
        toks[mat][row] = inputs[(size_t)(m_base + row) * SEQ + t];
    }
    __syncthreads();

    // gather 2 x 16 embedding rows of 128 floats = 1024 float4 transfers
    for (int i = tid; i < 1024; i += 256) {
        const int mat = i >> 9;
        const int rem = i & 511;
        const int row = rem >> 5;
        const int seg = rem & 31;
        const int tok = toks[mat][row];
        const float4 v = ((const float4*)weight)[(size_t)tok * (EMB / 4) + seg];
        *(float4*)&Xs[mat][row][seg * 4] = v;
    }
    __syncthreads();

    const int wave = tid >> 5;
    const int lane = tid & 31;
    const int half = lane >> 4;   // 0: lanes 0-15, 1: lanes 16-31
    const int l16  = lane & 15;
    const int koff = half * 2;    // K sub-offset held by this half-wave (A and B)

    #pragma unroll
    for (int j = 0; j < 2; ++j) {
        const int n0 = wave * 32 + j * 16;      // hy column base for this tile
        v8f hacc = {};

        for (int dir = 0; dir < 2; ++dir) {
            const float* __restrict__ Wx = dir ? Wxb : Wxf;
            v8f acc_i = {}, acc_g = {}, acc_o = {};

            #pragma unroll 4
            for (int ks = 0; ks < EMB / 4; ++ks) {
                const int k0 = ks * 4;
                // A fragment (16x4 f32): lane -> row l16, VGPRs -> K = koff, koff+1
                v2f a;
                a.x = Xs[dir][l16][k0 + koff];
                a.y = Xs[dir][l16][k0 + koff + 1];
                // B fragments (4x16 f32) for gates i (0), g (2H), o (3H):
                // B[k][n] = Wx[(goff + n0 + n) * EMB + k]
                const float* pi = Wx + (size_t)(0 * HID + n0 + l16) * EMB + k0 + koff;
                const float* pg = Wx + (size_t)(2 * HID + n0 + l16) * EMB + k0 + koff;
                const float* po = Wx + (size_t)(3 * HID + n0 + l16) * EMB + k0 + koff;
                v2f bi_f = { pi[0], pi[1] };
                v2f bg_f = { pg[0], pg[1] };
                v2f bo_f = { po[0], po[1] };

                acc_i = __builtin_amdgcn_wmma_f32_16x16x4_f32(false, a, false, bi_f,
                                                              (short)0, acc_i, false, false);
                acc_g = __builtin_amdgcn_wmma_f32_16x16x4_f32(false, a, false, bg_f,
                                                              (short)0, acc_g, false, false);
                acc_o = __builtin_amdgcn_wmma_f32_16x16x4_f32(false, a, false, bo_f,
                                                              (short)0, acc_o, false, false);
            }

            // biases: gates see bx + bh (h2h(0) contributes only its bias)
            const float* __restrict__ bx = dir ? bxb : bxf;
            const float* __restrict__ bh = dir ? bhb : bhf;
            const float bi = bx[0 * HID + n0 + l16] + bh[0 * HID + n0 + l16];
            const float bg = bx[2 * HID + n0 + l16] + bh[2 * HID + n0 + l16];
            const float bo = bx[3 * HID + n0 + l16] + bh[3 * HID + n0 + l16];

            #pragma unroll
            for (int r = 0; r < 8; ++r) {
                const float iv = acc_i[r] + bi;
                const float gv = acc_g[r] + bg;
                const float ov = acc_o[r] + bo;
                const float cy = sigmoidf_(iv) * fast_tanh(gv);  // c=0: forget term dead
                hacc[r] += sigmoidf_(ov) * fast_tanh(cy);
            }
        }

        // C/D layout: VGPR r holds row (r + 8*half), column l16
        #pragma unroll
        for (int r = 0; r < 8; ++r) {
            const int row = m_base + r + half * 8;
            hy[(size_t)row * HID + n0 + l16] = hacc[r];
        }
    }
}

// ---------------------------------------------------------------------------
// Kernel 2: logits = hy @ Why.T + by via WMMA, then row-wise log_softmax.
// One block, 8 waves; wave w owns rows [16w, 16w+16), both 16-col tiles.
// ---------------------------------------------------------------------------
__global__ __launch_bounds__(256) void bilstm_head_kernel(
    const float* __restrict__ hy,
    const float* __restrict__ Why,
    const float* __restrict__ by,
    float* __restrict__ out)
{
    const int tid  = threadIdx.x;
    const int wave = tid >> 5;
    const int lane = tid & 31;
    const int half = lane >> 4;
    const int l16  = lane & 15;
    const int koff = half * 2;
    const int m0   = wave * 16;

    v8f acc0 = {}, acc1 = {};
    #pragma unroll 4
    for (int ks = 0; ks < HID / 4; ++ks) {
        const int k0 = ks * 4;
        const float* pa = hy + (size_t)(m0 + l16) * HID + k0 + koff;
        v2f a = { pa[0], pa[1] };
        const float* p0 = Why + (size_t)(l16) * HID + k0 + koff;        // cols 0..15
        const float* p1 = Why + (size_t)(16 + l16) * HID + k0 + koff;   // cols 16..31
        v2f b0 = { p0[0], p0[1] };
        v2f b1 = { p1[0], p1[1] };
        acc0 = __builtin_amdgcn_wmma_f32_16x16x4_f32(false, a, false, b0,
                                                     (short)0, acc0, false, false);
        acc1 = __builtin_amdgcn_wmma_f32_16x16x4_f32(false, a, false, b1,
                                                     (short)0, acc1, false, false);
    }

    const float by0 = by[l16];
    const float by1 = by[16 + l16];

    #pragma unroll
    for (int r = 0; r < 8; ++r) {
        float x0 = acc0[r] + by0;
        float x1 = acc1[r] + by1;
        // row r+8*half lives across the 16 lanes of this half-wave (cols via l16);
        // xor masks 1/2/4/8 stay inside each 16-lane half on wave32.
        float m = fmaxf(x0, x1);
        #pragma unroll
        for (int mask = 1; mask < 16; mask <<= 1)
            m = fmaxf(m, __shfl_xor(m, mask, 32));
        float s = __expf(x0 - m) + __expf(x1 - m);
        #pragma unroll
        for (int mask = 1; mask < 16; mask <<= 1)
            s += __shfl_xor(s, mask, 32);
        const float lse = m + __logf(s);
        const int row = m0 + r + half * 8;
        out[(size_t)row * NOUT + l16]      = x0 - lse;
        out[(size_t)row * NOUT + 16 + l16] = x1 - lse;
    }
}

// ---------------------------------------------------------------------------
extern "C" void kernel_launch(void* const* d_in, const int* in_sizes, int n_in,
                              void* d_out, int out_size, void* d_ws, size_t ws_size,
                              hipStream_t stream)
{
    (void)in_sizes; (void)n_in; (void)out_size; (void)ws_size;

    const int*   inputs = (const int*)  d_in[0];
    const float* weight = (const float*)d_in[1];
    const float* Wxf    = (const float*)d_in[2];
    const float* bxf    = (const float*)d_in[3];
    // d_in[4] = Whf: dead (h=0 -> only bias bhf survives)
    const float* bhf    = (const float*)d_in[5];
    const float* Wxb    = (const float*)d_in[6];
    const float* bxb    = (const float*)d_in[7];
    // d_in[8] = Whb: dead
    const float* bhb    = (const float*)d_in[9];
    const float* Why    = (const float*)d_in[10];
    const float* by     = (const float*)d_in[11];

    float* out = (float*)d_out;
    float* hy  = (float*)d_ws;   // 128*256 f32 = 128 KB intermediate

    bilstm_gates_kernel<<<8, 256, 0, stream>>>(inputs, weight,
                                               Wxf, bxf, bhf,
                                               Wxb, bxb, bhb, hy);
    bilstm_head_kernel<<<1, 256, 0, stream>>>(hy, Why, by, out);
}